// GraphNetDecoder_34660386079056
// MI455X (gfx1250) — compile-verified
//
#include <hip/hip_runtime.h>
#include <math.h>

// Problem sizes (match reference)
constexpr int kB = 1024;
constexpr int kN = 4096;
constexpr int kH = 32;
constexpr int kKPad = 68;   // 2H+1 = 65, padded to multiple of 4 for WMMA K-steps

typedef __attribute__((ext_vector_type(2))) float v2f;
typedef __attribute__((ext_vector_type(8))) float v8f;

// ---------------------------------------------------------------------------
// Kernel 1: per-batch mean over N + gather context node -> cat[b, 0:68]
//   cat[b, 0:32]  = mean_n ne[b,n,:]
//   cat[b, 32:64] = ne[b, idxc[b], :]
//   cat[b, 64]    = times[b];  cat[b, 65:68] = 0 (pad)
// One block per batch, 256 threads, float4-coalesced streaming (512 KB/block).
// ---------------------------------------------------------------------------
__global__ __launch_bounds__(256) void k_reduce_cat(
    const float* __restrict__ ne, const float* __restrict__ times,
    const int* __restrict__ idxc, float* __restrict__ cat)
{
  __shared__ float part[32][32];          // [row-group][h]
  const int b   = blockIdx.x;
  const int tid = threadIdx.x;
  const int q   = tid & 7;                // which float4 within a 32-float row
  const int g   = tid >> 3;               // row-group 0..31
  const float* base = ne + (size_t)b * kN * kH;
  const float4* p = (const float4*)base;

  float4 acc = make_float4(0.f, 0.f, 0.f, 0.f);
  // rows n = g + 32*i ; float4 index = n*8 + q
  for (int i = 0; i < kN / 32; ++i) {
    float4 v = p[(size_t)(g * 8 + q) + (size_t)i * 256];
    acc.x += v.x; acc.y += v.y; acc.z += v.z; acc.w += v.w;
  }
  part[g][q * 4 + 0] = acc.x;
  part[g][q * 4 + 1] = acc.y;
  part[g][q * 4 + 2] = acc.z;
  part[g][q * 4 + 3] = acc.w;
  __syncthreads();

  if (tid < 32) {                         // tid == h
    float s = 0.f;
    #pragma unroll
    for (int gg = 0; gg < 32; ++gg) s += part[gg][tid];
    cat[(size_t)b * kKPad + tid] = s * (1.0f / (float)kN);     // node_avg
    int ic = idxc[b];
    cat[(size_t)b * kKPad + 32 + tid] = base[(size_t)ic * kH + tid];  // f
  }
  if (tid >= 64 && tid < kKPad) {
    cat[(size_t)b * kKPad + tid] = (tid == 64) ? times[b] : 0.0f;
  }
}

// ---------------------------------------------------------------------------
// Kernel 2: qk[B,32] via fp32 WMMA chain
//   emb = cat @ W1^T + b1 ; q = emb @ Wq^T ; qk = q @ Wk
// Grid: B/16 blocks of 64 threads (2 waves). Wave w owns output columns
// [16w, 16w+16). V_WMMA_F32_16X16X4_F32, K stepped by 4, intermediates in LDS.
// A layout: lanes 0-15 hold (K=k0,k0+1) of row M=lane; lanes 16-31 hold
// (K=k0+2,k0+3) of row M=lane-16. B/C/D: row striped across lanes.
// ---------------------------------------------------------------------------
__global__ __launch_bounds__(64) void k_qk_wmma(
    const float* __restrict__ cat, const float* __restrict__ W1,
    const float* __restrict__ b1,  const float* __restrict__ Wq,
    const float* __restrict__ Wk,  float* __restrict__ qk)
{
  __shared__ float embT[16][32];
  __shared__ float qT[16][32];

  const int tile    = blockIdx.x;              // rows tile*16 .. tile*16+15
  const int lane    = threadIdx.x & 31;
  const int wave    = threadIdx.x >> 5;        // N-tile (0 or 1)
  const int halfsel = lane >> 4;               // 0: lanes 0-15, 1: lanes 16-31
  const int l15     = lane & 15;
  const int rowA    = tile * 16 + l15;         // A-matrix row for this lane
  const int colB    = wave * 16 + l15;         // B/C/D column for this lane

  // ---- GEMM1: emb = cat @ W1^T, K = 68 (65 real + zero pad) ----
  v8f c1 = {};
  for (int k0 = 0; k0 < kKPad; k0 += 4) {
    int ka = k0 + halfsel * 2;
    v2f a;  a.x = cat[(size_t)rowA * kKPad + ka];
            a.y = cat[(size_t)rowA * kKPad + ka + 1];
    v2f bm; bm.x = (ka     < 65) ? W1[colB * 65 + ka    ] : 0.0f;   // W1^T[k][n]
            bm.y = (ka + 1 < 65) ? W1[colB * 65 + ka + 1] : 0.0f;
    c1 = __builtin_amdgcn_wmma_f32_16x16x4_f32(false, a, false, bm,
                                               (short)0, c1, false, false);
  }
  float bias = b1[colB];
  #pragma unroll
  for (int r = 0; r < 8; ++r) {
    int m = r + halfsel * 8;                  // C layout: VGPR r -> M = r (+8)
    embT[m][colB] = c1[r] + bias;
  }
  __syncthreads();

  // ---- GEMM2: q = emb @ Wq^T, K = 32 ----
  v8f c2 = {};
  for (int k0 = 0; k0 < kH; k0 += 4) {
    int ka = k0 + halfsel * 2;
    v2f a;  a.x = embT[l15][ka];  a.y = embT[l15][ka + 1];
    v2f bm; bm.x = Wq[colB * kH + ka];       // Wq^T[k][n] = Wq[n][k]
            bm.y = Wq[colB * kH + ka + 1];
    c2 = __builtin_amdgcn_wmma_f32_16x16x4_f32(false, a, false, bm,
                                               (short)0, c2, false, false);
  }
  #pragma unroll
  for (int r = 0; r < 8; ++r) {
    int m = r + halfsel * 8;
    qT[m][colB] = c2[r];
  }
  __syncthreads();

  // ---- GEMM3: qk = q @ Wk, K = 32 ----
  v8f c3 = {};
  for (int k0 = 0; k0 < kH; k0 += 4) {
    int ka = k0 + halfsel * 2;
    v2f a;  a.x = qT[l15][ka];  a.y = qT[l15][ka + 1];
    v2f bm; bm.x = Wk[ka * kH + colB];       // Wk[k][n]
            bm.y = Wk[(ka + 1) * kH + colB];
    c3 = __builtin_amdgcn_wmma_f32_16x16x4_f32(false, a, false, bm,
                                               (short)0, c3, false, false);
  }
  #pragma unroll
  for (int r = 0; r < 8; ++r) {
    int m = r + halfsel * 8;
    qk[(size_t)(tile * 16 + m) * kH + colB] = c3[r];
  }
}

// ---------------------------------------------------------------------------
// Kernel 3: scores + softmax. One block per batch, 256 threads.
//   scores[b,n] = (qk[b] . ne[b,n]) / sqrt(H) - mask[b,n]*1e9 ; softmax over n
// Streams 512 KB/block of ne via float4; scores staged in 16 KB LDS.
// ---------------------------------------------------------------------------
__global__ __launch_bounds__(256) void k_scores_softmax(
    const float* __restrict__ ne, const float* __restrict__ mask,
    const float* __restrict__ qk, float* __restrict__ out)
{
  __shared__ float sc[kN];       // 16 KB
  __shared__ float red[256];
  const int b   = blockIdx.x;
  const int tid = threadIdx.x;
  const float* base = ne + (size_t)b * kN * kH;

  float4 qv[8];
  const float4* qp = (const float4*)(qk + (size_t)b * kH);
  #pragma unroll
  for (int j = 0; j < 8; ++j) qv[j] = qp[j];

  const float scale = 0.17677669529663687f;   // 1/sqrt(32)
  float lmax = -INFINITY;
  for (int i = 0; i < kN / 256; ++i) {
    int n = tid + 256 * i;
    const float4* rp = (const float4*)(base + (size_t)n * kH);
    float d = 0.f;
    #pragma unroll
    for (int j = 0; j < 8; ++j) {
      float4 v = rp[j];
      d += v.x * qv[j].x + v.y * qv[j].y + v.z * qv[j].z + v.w * qv[j].w;
    }
    float s = d * scale - mask[(size_t)b * kN + n] * 999999999.0f;
    sc[n] = s;
    lmax = fmaxf(lmax, s);
  }
  red[tid] = lmax;
  __syncthreads();
  for (int off = 128; off > 0; off >>= 1) {
    if (tid < off) red[tid] = fmaxf(red[tid], red[tid + off]);
    __syncthreads();
  }
  float mx = red[0];
  __syncthreads();

  float lsum = 0.f;
  for (int i = 0; i < kN / 256; ++i) {
    int n = tid + 256 * i;
    float e = __expf(sc[n] - mx);
    sc[n] = e;
    lsum += e;
  }
  red[tid] = lsum;
  __syncthreads();
  for (int off = 128; off > 0; off >>= 1) {
    if (tid < off) red[tid] += red[tid + off];
    __syncthreads();
  }
  float inv = 1.0f / red[0];

  for (int i = 0; i < kN / 256; ++i) {
    int n = tid + 256 * i;
    out[(size_t)b * kN + n] = sc[n] * inv;
  }
}

// ---------------------------------------------------------------------------
extern "C" void kernel_launch(void* const* d_in, const int* in_sizes, int n_in,
                              void* d_out, int out_size, void* d_ws, size_t ws_size,
                              hipStream_t stream) {
  (void)in_sizes; (void)n_in; (void)out_size; (void)ws_size;
  const float* ne    = (const float*)d_in[0];   // [B,N,H]
  const float* mask  = (const float*)d_in[1];   // [B,N]
  const float* times = (const float*)d_in[2];   // [B]
  /* d_in[3] = vf : unused in idxc path */
  const float* W1    = (const float*)d_in[4];   // [H, 2H+1]
  const float* b1    = (const float*)d_in[5];   // [H]
  const float* Wq    = (const float*)d_in[6];   // [H,H]
  const float* Wk    = (const float*)d_in[7];   // [H,H]
  const int*   idxc  = (const int*)d_in[8];     // [B,1]
  float* out = (float*)d_out;                   // [B,N]

  float* cat = (float*)d_ws;                    // [B, 68]
  float* qk  = cat + (size_t)kB * kKPad;        // [B, 32]

  k_reduce_cat<<<kB, 256, 0, stream>>>(ne, times, idxc, cat);
  k_qk_wmma<<<kB / 16, 64, 0, stream>>>(cat, W1, b1, Wq, Wk, qk);
  k_scores_softmax<<<kB, 256, 0, stream>>>(ne, mask, qk, out);
}